// Attention_80573586473568
// MI455X (gfx1250) — compile-verified
//
#include <hip/hip_runtime.h>
#include <cstddef>
#include <cstdint>

typedef float v2f __attribute__((ext_vector_type(2)));
typedef float v8f __attribute__((ext_vector_type(8)));

constexpr int B     = 8;
constexpr int NQ    = 2048;
constexpr int NK    = 2048;
constexpr int DH    = 64;
constexpr int QT    = 16;            // q rows per workgroup
constexpr int WAVES = 8;             // 256 threads, wave32
constexpr int KC_W  = NK / WAVES;    // 256 k columns per wave
constexpr int KT_W  = KC_W / 16;     // 16 k tiles of 16 per wave
constexpr float EPS_C  = 1e-9f;
constexpr float SCALE  = 0.125f;     // 1/sqrt(64)

// LDS layout (dynamic shared; ~205 KB < 320 KB WGP limit).
// Row strides padded (+4 floats) so 16 lanes reading a column hit distinct banks.
struct Smem {
  float q_s[QT][68];            // Q tile, 16 x 64 (+4 pad)
  float k_s[WAVES][2][16][68];  // per-wave K staging, double-buffered
  float t_s[QT][NK + 4];        // final t for the PV matmul
  float red[WAVES][16];         // cross-wave row-reduction partials
  float stat[16];               // combined per-row statistic
  float qmf[16];                // q_mask as float per row
  float partial[4][16][16];     // split-K partial f tiles
};

// ---- gfx1250 async global->LDS DMA (ASYNCcnt-tracked) ----
// the vdst VGPR supplies the LDS address; generic shared pointers carry the LDS
// offset in their low 32 bits (shared aperture, flat addressing rules).
__device__ __forceinline__ void async_g2l_b128(const void* gaddr, const float* ldsaddr) {
  const uint32_t loff = (uint32_t)(uintptr_t)ldsaddr;
  asm volatile("global_load_async_to_lds_b128 %0, %1, off"
               :: "v"(loff), "v"(gaddr) : "memory");
}
__device__ __forceinline__ void async_wait0() {
  asm volatile("s_wait_asynccnt 0x0" ::: "memory");
}
__device__ __forceinline__ void async_wait8() {
  asm volatile("s_wait_asynccnt 0x8" ::: "memory");
}

__device__ __forceinline__ float hmax16(float v) {
  v = fmaxf(v, __shfl_xor(v, 1, 32));
  v = fmaxf(v, __shfl_xor(v, 2, 32));
  v = fmaxf(v, __shfl_xor(v, 4, 32));
  v = fmaxf(v, __shfl_xor(v, 8, 32));
  return v;
}
__device__ __forceinline__ float hsum16(float v) {
  v += __shfl_xor(v, 1, 32);
  v += __shfl_xor(v, 2, 32);
  v += __shfl_xor(v, 4, 32);
  v += __shfl_xor(v, 8, 32);
  return v;
}

__global__ void __launch_bounds__(256, 1)
attn_tau3_kernel(const float* __restrict__ qg, const float* __restrict__ kg,
                 const float* __restrict__ vg, const int* __restrict__ k_mask,
                 const int* __restrict__ q_mask, float* __restrict__ out_f,
                 float* __restrict__ out_t) {
  extern __shared__ char smem_raw[];
  Smem& sm = *reinterpret_cast<Smem*>(smem_raw);

  const int tid  = threadIdx.x;
  const int wave = tid >> 5;
  const int lane = tid & 31;
  const int h    = lane >> 4;   // half-wave: rows 8h..8h+7 of each C fragment
  const int ln   = lane & 15;   // column (B/C/D) or row (A) index within tile

  const int bidx = blockIdx.x;
  const int b    = bidx >> 7;         // / (NQ/QT) = /128
  const int q0   = (bidx & 127) * QT;

  // ---- Stage Q tile (16x64) via async DMA; q_mask via plain loads ----
  {
    const int row = tid >> 4, c4 = tid & 15;   // 256 lanes -> 256 float4
    async_g2l_b128(&qg[((size_t)b * NQ + q0 + row) * DH + c4 * 4],
                   &sm.q_s[row][c4 * 4]);
  }
  if (tid < 16) sm.qmf[tid] = (float)q_mask[(size_t)b * NQ + q0 + tid];
  // warm L2 for this batch's V (used in phase 6)
  __builtin_prefetch(&vg[(size_t)b * NK * DH + (size_t)tid * 512], 0, 2);
  async_wait0();          // each wave's own async stores to LDS complete
  __syncthreads();        // cross-wave visibility of q_s

  // ---- Phase 1: S = (Q K^T) * scale; double-buffered async K staging ----
  const int n_base = wave * KC_W;
  const float* kwbase = kg + ((size_t)b * NK + n_base) * DH;
  const int st_row = lane >> 1, st_c4 = (lane & 1) * 8;  // unused helper vars

  // issue tile 0 into buffer 0 (8 x B128 per wave = 16x64 floats)
#pragma unroll
  for (int i = 0; i < 8; ++i) {
    int f4 = i * 32 + lane;
    int row = f4 >> 4, c4 = f4 & 15;
    async_g2l_b128(kwbase + (size_t)row * DH + c4 * 4,
                   &sm.k_s[wave][0][row][c4 * 4]);
  }

  v8f S[KT_W];
#pragma unroll
  for (int kt = 0; kt < KT_W; ++kt) {
    // issue next tile into the other buffer while computing this one
    if (kt + 1 < KT_W) {
      const float* knext = kwbase + (size_t)(kt + 1) * 16 * DH;
#pragma unroll
      for (int i = 0; i < 8; ++i) {
        int f4 = i * 32 + lane;
        int row = f4 >> 4, c4 = f4 & 15;
        async_g2l_b128(knext + (size_t)row * DH + c4 * 4,
                       &sm.k_s[wave][(kt + 1) & 1][row][c4 * 4]);
      }
      async_wait8();   // tile kt resident (async loads complete in order)
    } else {
      async_wait0();
    }

    const float* ks = &sm.k_s[wave][kt & 1][0][0];
    v8f c = {};
    for (int kk = 0; kk < 16; ++kk) {          // K = 64 in steps of 4
      const int kb = kk * 4 + 2 * h;
      v2f a;  a.x  = sm.q_s[ln][kb];     a.y  = sm.q_s[ln][kb + 1];
      v2f bb; bb.x = ks[ln * 68 + kb];   bb.y = ks[ln * 68 + kb + 1];
      c = __builtin_amdgcn_wmma_f32_16x16x4_f32(false, a, false, bb,
                                                (short)0, c, false, false);
    }
#pragma unroll
    for (int r = 0; r < 8; ++r) S[kt][r] = c[r] * SCALE;
  }

  // ---- Phase 2: row max (over raw scores, like the reference) ----
  v8f mx = S[0];
#pragma unroll
  for (int kt = 1; kt < KT_W; ++kt)
#pragma unroll
    for (int r = 0; r < 8; ++r) mx[r] = fmaxf(mx[r], S[kt][r]);

  float rowmax[8];
#pragma unroll
  for (int r = 0; r < 8; ++r) rowmax[r] = hmax16(mx[r]);
  if (ln == 0) {
#pragma unroll
    for (int r = 0; r < 8; ++r) sm.red[wave][8 * h + r] = rowmax[r];
  }
  __syncthreads();
  if (tid < 16) {
    float m = sm.red[0][tid];
#pragma unroll
    for (int w2 = 1; w2 < WAVES; ++w2) m = fmaxf(m, sm.red[w2][tid]);
    sm.stat[tid] = m;
  }
  __syncthreads();
#pragma unroll
  for (int r = 0; r < 8; ++r) rowmax[r] = sm.stat[8 * h + r];

  // ---- Phase 3: e = exp(w - max) * mask ; rowsum ----
  float kmf[KT_W];
#pragma unroll
  for (int kt = 0; kt < KT_W; ++kt)
    kmf[kt] = (float)k_mask[(size_t)b * NK + n_base + kt * 16 + ln];
  float qmr[8];
#pragma unroll
  for (int r = 0; r < 8; ++r) qmr[r] = sm.qmf[8 * h + r];

  v8f ssum = {};
#pragma unroll
  for (int kt = 0; kt < KT_W; ++kt)
#pragma unroll
    for (int r = 0; r < 8; ++r) {
      float e = expf(S[kt][r] - rowmax[r]) * (kmf[kt] * qmr[r]);
      S[kt][r] = e;
      ssum[r] += e;
    }

  float rowsum[8];
#pragma unroll
  for (int r = 0; r < 8; ++r) rowsum[r] = hsum16(ssum[r]);
  if (ln == 0) {
#pragma unroll
    for (int r = 0; r < 8; ++r) sm.red[wave][8 * h + r] = rowsum[r];
  }
  __syncthreads();
  if (tid < 16) {
    float s = sm.red[0][tid];
#pragma unroll
    for (int w2 = 1; w2 < WAVES; ++w2) s += sm.red[w2][tid];
    sm.stat[tid] = s;
  }
  __syncthreads();
#pragma unroll
  for (int r = 0; r < 8; ++r) rowsum[r] = sm.stat[8 * h + r] + EPS_C;

  // ---- Phase 4: t1 = e/(sum+EPS); tp = t1^3 ; rowsum2 ----
  v8f ssum2 = {};
#pragma unroll
  for (int kt = 0; kt < KT_W; ++kt)
#pragma unroll
    for (int r = 0; r < 8; ++r) {
      float t1 = S[kt][r] / rowsum[r];
      float tp = t1 * t1 * t1;
      S[kt][r] = tp;
      ssum2[r] += tp;
    }

  float rowsum2[8];
#pragma unroll
  for (int r = 0; r < 8; ++r) rowsum2[r] = hsum16(ssum2[r]);
  if (ln == 0) {
#pragma unroll
    for (int r = 0; r < 8; ++r) sm.red[wave][8 * h + r] = rowsum2[r];
  }
  __syncthreads();
  if (tid < 16) {
    float s = sm.red[0][tid];
#pragma unroll
    for (int w2 = 1; w2 < WAVES; ++w2) s += sm.red[w2][tid];
    sm.stat[tid] = s;
  }
  __syncthreads();
#pragma unroll
  for (int r = 0; r < 8; ++r) rowsum2[r] = sm.stat[8 * h + r];
  // (all-masked rows: 0/0 -> NaN, same as the reference)

  // ---- Phase 5: final t -> global + LDS ----
#pragma unroll
  for (int kt = 0; kt < KT_W; ++kt) {
    const int col = n_base + kt * 16 + ln;
#pragma unroll
    for (int r = 0; r < 8; ++r) {
      const int row = 8 * h + r;
      const float tv = S[kt][r] / rowsum2[r];
      out_t[((size_t)b * NQ + q0 + row) * NK + col] = tv;
      sm.t_s[row][col] = tv;
    }
  }
  __syncthreads();

  // ---- Phase 6: f = t @ V, split-K across wave pairs ----
  const int dt = wave & 3;   // which 16-col slice of D
  const int kh = wave >> 2;  // which half of K
  const int dcol = dt * 16 + ln;
  const float* vbase = vg + (size_t)b * NK * DH + dcol;

  v8f cf = {};
  for (int ks = 0; ks < 256; ++ks) {
    const int k0 = kh * 1024 + ks * 4 + 2 * h;
    v2f a;  a.x  = sm.t_s[ln][k0];           a.y  = sm.t_s[ln][k0 + 1];
    v2f bb; bb.x = vbase[(size_t)k0 * DH];   bb.y = vbase[(size_t)(k0 + 1) * DH];
    cf = __builtin_amdgcn_wmma_f32_16x16x4_f32(false, a, false, bb,
                                               (short)0, cf, false, false);
  }
  if (kh == 1) {
#pragma unroll
    for (int r = 0; r < 8; ++r) sm.partial[dt][8 * h + r][ln] = cf[r];
  }
  __syncthreads();
  if (kh == 0) {
#pragma unroll
    for (int r = 0; r < 8; ++r) {
      const float val = cf[r] + sm.partial[dt][8 * h + r][ln];
      out_f[((size_t)b * NQ + q0 + 8 * h + r) * DH + dcol] = val;
    }
  }
}

extern "C" void kernel_launch(void* const* d_in, const int* in_sizes, int n_in,
                              void* d_out, int out_size, void* d_ws, size_t ws_size,
                              hipStream_t stream) {
  (void)in_sizes; (void)n_in; (void)out_size; (void)d_ws; (void)ws_size;
  const float* q      = (const float*)d_in[0];
  const float* k      = (const float*)d_in[1];
  const float* v      = (const float*)d_in[2];
  const int*   k_mask = (const int*)d_in[3];
  const int*   q_mask = (const int*)d_in[4];

  float* out_f = (float*)d_out;                       // [B, NQ, DH]
  float* out_t = out_f + (size_t)B * NQ * DH;         // [B, NQ, NK]

  const dim3 grid(B * (NQ / QT));  // 1024 workgroups
  const dim3 block(WAVES * 32);    // 8 wave32 waves
  attn_tau3_kernel<<<grid, block, sizeof(Smem), stream>>>(q, k, v, k_mask,
                                                          q_mask, out_f, out_t);
}